// CustomLoss_54400055771232
// MI455X (gfx1250) — compile-verified
//
#include <hip/hip_runtime.h>

typedef __attribute__((ext_vector_type(2))) float f32x2;
typedef __attribute__((ext_vector_type(8))) float f32x8;

#define NBLK 4096

// Full 32-lane sum using V_WMMA_F32_16X16X4_F32.
// A layout (32-bit A 16x4): lane L, VGPR0 -> A[L%16][L<16?0:2]; VGPR1 -> A[L%16][L<16?1:3].
// Put v in VGPR0, 0 in VGPR1; B = all ones => D[m][n] = v[m] + v[m+16].
// C/D layout: VGPR j, lane L -> D[j + (L<16?0:8)][L%16]. Sum of 8 D VGPRs per lane:
// lanes 0-15 get sum(t[0..7]), lanes 16-31 get sum(t[8..15]); xor-shuffle by 16 totals it.
__device__ __forceinline__ float wave_sum32_wmma(float v) {
  f32x2 a; a[0] = v;    a[1] = 0.0f;
  f32x2 b; b[0] = 1.0f; b[1] = 1.0f;
  f32x8 c = {};
  c = __builtin_amdgcn_wmma_f32_16x16x4_f32(false, a, false, b, (short)0, c, false, false);
  float s = c[0] + c[1] + c[2] + c[3] + c[4] + c[5] + c[6] + c[7];
  s += __shfl_xor(s, 16, 32);
  return s;
}

__global__ void __launch_bounds__(256)
loss_partial_kernel(const float* __restrict__ preds,
                    const float* __restrict__ targets,
                    const float* __restrict__ nee_qc,
                    const int*   __restrict__ igbp,
                    const int*   __restrict__ koppen,
                    const float* __restrict__ igbp_table,
                    const float* __restrict__ koppen_table,
                    float* __restrict__ partials,
                    int N) {
  __shared__ float s_igbp[16];
  __shared__ float s_kop[8];
  __shared__ float s_wave[8];
  const int tid = threadIdx.x;
  if (tid < 16) s_igbp[tid] = igbp_table[tid];
  if (tid < 5)  s_kop[tid]  = koppen_table[tid];
  __syncthreads();

  float acc = 0.0f;
  const int stride = NBLK * 256;
  for (int i = blockIdx.x * 256 + tid; i < N; i += stride) {
    const size_t base = (size_t)i * 6u;
    // 6 floats per site at 24B stride: three 8B-aligned b64 loads, non-temporal
    // (360MB stream > 192MB L2; keep the weight tables resident instead).
    f32x2 p01 = __builtin_nontemporal_load((const f32x2*)(preds + base));
    f32x2 p23 = __builtin_nontemporal_load((const f32x2*)(preds + base + 2));
    f32x2 p45 = __builtin_nontemporal_load((const f32x2*)(preds + base + 4));
    f32x2 t01 = __builtin_nontemporal_load((const f32x2*)(targets + base));
    f32x2 t23 = __builtin_nontemporal_load((const f32x2*)(targets + base + 2));
    f32x2 t45 = __builtin_nontemporal_load((const f32x2*)(targets + base + 4));
    float qc  = __builtin_nontemporal_load(nee_qc + i);
    int   ig  = __builtin_nontemporal_load(igbp + i);
    int   ko  = __builtin_nontemporal_load(koppen + i);

    float d0 = p01[0] - t01[0], d1 = p01[1] - t01[1];
    float d2 = p23[0] - t23[0], d3 = p23[1] - t23[1];
    float d4 = p45[0] - t45[0], d5 = p45[1] - t45[1];
    float mse = (d0*d0 + d1*d1 + d2*d2 + d3*d3 + d4*d4 + d5*d5) * (1.0f / 6.0f);
    float w   = qc * s_igbp[ig & 15] * s_kop[ko & 7];
    // balance_error = (p2 - (-(p0 - p1)))^2 = (p2 + p0 - p1)^2
    float bal = p23[0] + p01[0] - p01[1];
    acc = fmaf(0.1f, bal * bal, fmaf(mse, w, acc));
  }

  // intra-wave reduction via WMMA (EXEC all-1s here: full waves, reconverged)
  float ws = wave_sum32_wmma(acc);
  const int lane = tid & 31, wave = tid >> 5;
  if (lane == 0) s_wave[wave] = ws;
  __syncthreads();
  if (tid == 0) {
    float s = 0.0f;
#pragma unroll
    for (int w = 0; w < 8; ++w) s += s_wave[w];
    partials[blockIdx.x] = s;
  }
}

__global__ void __launch_bounds__(256)
final_reduce_kernel(const float* __restrict__ partials, int n, float inv_total,
                    float* __restrict__ out) {
  __shared__ float s_wave[8];
  const int tid = threadIdx.x;
  float acc = 0.0f;
  for (int i = tid; i < n; i += 256) acc += partials[i];
  float ws = wave_sum32_wmma(acc);
  if ((tid & 31) == 0) s_wave[tid >> 5] = ws;
  __syncthreads();
  if (tid == 0) {
    float s = 0.0f;
#pragma unroll
    for (int w = 0; w < 8; ++w) s += s_wave[w];
    out[0] = s * inv_total;
  }
}

extern "C" void kernel_launch(void* const* d_in, const int* in_sizes, int n_in,
                              void* d_out, int out_size, void* d_ws, size_t ws_size,
                              hipStream_t stream) {
  const float* preds        = (const float*)d_in[0];
  const float* targets      = (const float*)d_in[1];
  const float* nee_qc       = (const float*)d_in[2];
  const int*   igbp         = (const int*)  d_in[3];
  const int*   koppen       = (const int*)  d_in[4];
  const float* igbp_table   = (const float*)d_in[5];
  const float* koppen_table = (const float*)d_in[6];
  float*       out          = (float*)d_out;

  const int N = in_sizes[2];           // B*T (nee_qc flat count)
  const float inv_total = 1.0f / (float)N;
  float* partials = (float*)d_ws;      // NBLK floats of scratch

  loss_partial_kernel<<<NBLK, 256, 0, stream>>>(preds, targets, nee_qc, igbp, koppen,
                                                igbp_table, koppen_table, partials, N);
  final_reduce_kernel<<<1, 256, 0, stream>>>(partials, NBLK, inv_total, out);
}